// TnoCausalV4_24747601559865
// MI455X (gfx1250) — compile-verified
//
#include <hip/hip_runtime.h>

// Depthwise causal Toeplitz matvec, (b,n,d) = (4,4096,1024), fp32.
//   y[b,i,d] = sum_{j<=i} t[i-j,d] * x[b,j,d]
//
// One block per channel d. LDS holds two reversed copies of t[:,d] (merged in
// one array so the per-lane parity pick is a 32-bit offset) and x[b,:,d] with
// a 48-element zero pad in front of each batch column, which makes every
// k >= 1 iteration mask-free (boundary J<0 reads hit zeros). Each wave
// carries 8 accumulators so one A=T(k) fetch (aligned v2f pairs) feeds up to
// 32 v_wmma_f32_16x16x4_f32 per k. x staging uses the Tensor Data Mover
// (tensor_load_to_lds, one 2D column-gather descriptor per batch) completed
// with s_wait_tensorcnt.

typedef __attribute__((ext_vector_type(2))) float v2f;
typedef __attribute__((ext_vector_type(8))) float v8f;
typedef __attribute__((ext_vector_type(4))) unsigned int uint32x4;
typedef __attribute__((ext_vector_type(8))) int int32x8;
typedef __attribute__((ext_vector_type(4))) int int32x4;

#ifndef __has_builtin
#define __has_builtin(x) 0
#endif
#if __has_builtin(__builtin_amdgcn_tensor_load_to_lds) && \
    __has_builtin(__builtin_amdgcn_s_wait_tensorcnt)
#define USE_TDM 1
#else
#define USE_TDM 0
#endif

namespace {
constexpr int kB = 4;            // batches
constexpr int kN = 4096;         // sequence length
constexpr int kD = 1024;         // channels
constexpr int kC = 16;           // chunk (WMMA tile edge)
constexpr int kWaves = 8;        // waves per block
constexpr int NG = 8;            // accumulator groups per wave
constexpr int kPad = 48;         // zero pad in front of each x column (3 chunks)
constexpr int kNP = kN + kPad;   // padded per-batch stride in LDS (4144)
}

__device__ __forceinline__ v8f wmma_f32(v2f a, v2f b, v8f c) {
  return __builtin_amdgcn_wmma_f32_16x16x4_f32(
      /*neg_a=*/false, a, /*neg_b=*/false, b,
      /*c_mod=*/(short)0, c, /*reuse_a=*/false, /*reuse_b=*/false);
}

__global__ __launch_bounds__(256) void toeplitz_causal_wmma(
    const float* __restrict__ x,   // (kB, kN, kD)
    const float* __restrict__ t,   // (kN, kD)
    float* __restrict__ y) {       // (kB, kN, kD)
  // RR[0..kN)      = t[4095 - i]           (for odd-col lanes)
  // RR[kN..2kN+8)  = t[4096 - (i-kN)], 0 OOB (for even-col lanes)
  __shared__ float RR[2 * kN + 8];
  __shared__ float sXp[kB * kNP];  // per batch: [48 zeros][x[b, 0..4095, d]]

  const int d   = blockIdx.x;
  const int tid = threadIdx.x;
  const int wave = tid >> 5;

#if USE_TDM
  // ---- TDM: one 2D column-gather per batch, issued by wave 0.
  // Tile 1 x 4096, element 4B, row stride 1024 elements -> contiguous LDS.
  if (wave == 0) {
    const unsigned ldsBase = (unsigned)(uintptr_t)&sXp[0];
#pragma unroll
    for (int b = 0; b < kB; ++b) {
      const unsigned long long ga =
          (unsigned long long)(uintptr_t)(x + (size_t)b * kN * kD + d);
      uint32x4 g0;
      g0[0] = 1u;                                            // count=1 (valid)
      g0[1] = ldsBase + (unsigned)((b * kNP + kPad) * 4);    // lds_addr
      g0[2] = (unsigned)(ga & 0xffffffffu);                  // global_addr lo
      g0[3] = (unsigned)((ga >> 32) & 0x01ffffffu) | (2u << 30);  // hi | type=2
      int32x8 g1;
      g1[0] = (int)(2u << 16);      // data_size=2 (4 bytes)
      g1[1] = (int)(1u << 16);      // tensor_dim0 = 1 (lo16)
      g1[2] = (int)(4096u << 16);   // tensor_dim0 hi=0 | tensor_dim1 lo16=4096
      g1[3] = (int)(1u << 16);      // tensor_dim1 hi=0 | tile_dim0 = 1
      g1[4] = (int)(4096u);         // tile_dim1 = 4096 | tile_dim2 = 0 (2D)
      g1[5] = (int)(1024u);         // tensor_dim0_stride = 1024 elements (lo32)
      g1[6] = 0;                    // stride0 hi | stride1 lo (unused)
      g1[7] = 0;                    // stride1 hi (unused)
      int32x4 g2;
      g2[0] = 1; g2[1] = 1; g2[2] = 0; g2[3] = 0;  // dims 2/3 unused
      int32x4 g3;
      g3[0] = 0; g3[1] = 0; g3[2] = 0; g3[3] = 0;
#if defined(__clang_major__) && __clang_major__ >= 23
      int32x8 gz = {};
      __builtin_amdgcn_tensor_load_to_lds(g0, g1, g2, g3, gz, 0);
#else
      __builtin_amdgcn_tensor_load_to_lds(g0, g1, g2, g3, 0);
#endif
    }
  }
#else
#pragma unroll
  for (int b = 0; b < kB; ++b) {
    for (int i = tid; i < kN; i += 256) {
      sXp[b * kNP + kPad + i] = x[(b * kN + i) * kD + d];
    }
  }
#endif

  // ---- Stage reversed t copies and zero the x pads (all waves).
  for (int i = tid; i < kN; i += 256) {
    RR[i] = t[(kN - 1 - i) * kD + d];
  }
  for (int i = tid; i < kN + 8; i += 256) {
    const int src = kN - i;
    RR[kN + i] = (src >= 0 && src < kN) ? t[src * kD + d] : 0.0f;
  }
  for (int p = tid; p < kB * kPad; p += 256) {
    sXp[(p / kPad) * kNP + (p % kPad)] = 0.0f;
  }

#if USE_TDM
  if (wave == 0) __builtin_amdgcn_s_wait_tensorcnt(0);
#endif
  __syncthreads();

  const int lane  = tid & 31;
  const int col   = lane & 15;   // WMMA N column (== A-matrix row m)
  const int half  = lane >> 4;   // lane half -> K pair / M half
  const int batch = col & 3;
  const int gsub  = col >> 2;
  const int qoff  = half * 2;    // K_local = qoff + v

  // A pair for (k,s) lives at RR[aBase - 16k + 4s] (+1), 8B aligned by
  // construction: parity(ti0) == parity(col) selects which reversed copy.
  const int aBase = (col & 1) ? (kN - 1 - col + qoff)        // R0 region
                              : (2 * kN - col + qoff);       // R1 region

  int Ii_[NG];
  int bOff[NG];   // 32-bit LDS element offsets (no generic pointers)
#pragma unroll
  for (int j = 0; j < NG; ++j) {
    const int g = wave + kWaves * j;           // group id 0..63
    Ii_[j] = g * 4 + gsub;                     // this column's output chunk
    bOff[j] = batch * kNP + kPad + Ii_[j] * kC + qoff;
  }

  v8f acc[NG] = {};

  // ---- k == 0 peel: causal diagonal block (A masked; B clean).
  {
    v2f a[4];
#pragma unroll
    for (int s = 0; s < 4; ++s) {
#pragma unroll
      for (int v = 0; v < 2; ++v) {
        const int ti = col - (4 * s + qoff + v);
        const float tv = RR[(ti >= 0) ? (kN - 1 - ti) : 0];
        a[s][v] = (ti >= 0) ? tv : 0.0f;
      }
    }
#pragma unroll
    for (int j = 0; j < NG; ++j) {
#pragma unroll
      for (int s = 0; s < 4; ++s) {
        const v2f bb = *(const v2f*)(&sXp[0] + bOff[j] + 4 * s);
        acc[j] = wmma_f32(a[s], bb, acc[j]);
      }
    }
  }

  // ---- Phased sweep over k >= 1: zero masking anywhere (pad absorbs J<0).
  int aOff = aBase - kC;   // A pair base for current k
#pragma unroll
  for (int j = 0; j < NG; ++j) bOff[j] -= kC;

  int kCur = 1;
#pragma unroll
  for (int j = 0; j < NG; ++j) {
    const int kEnd = (wave + kWaves * j) * 4 + 3;   // last k for group j

    for (; kCur <= kEnd; ++kCur) {
      v2f a[4];
#pragma unroll
      for (int s = 0; s < 4; ++s) a[s] = *(const v2f*)(&RR[0] + aOff + 4 * s);
#pragma unroll
      for (int jj = j; jj < NG; ++jj) {
#pragma unroll
        for (int s = 0; s < 4; ++s) {
          const v2f bb = *(const v2f*)(&sXp[0] + bOff[jj] + 4 * s);
          acc[jj] = wmma_f32(a[s], bb, acc[jj]);
        }
      }
      aOff -= kC;
#pragma unroll
      for (int jj = j; jj < NG; ++jj) bOff[jj] -= kC;
    }

    // Store group j. D layout: VGPR r -> M = r + 8*half, N = col.
    const int ibase = Ii_[j] * kC + half * 8;
#pragma unroll
    for (int r = 0; r < 8; ++r) {
      __builtin_nontemporal_store(acc[j][r],
                                  &y[(batch * kN + (ibase + r)) * kD + d]);
    }
  }
}

extern "C" void kernel_launch(void* const* d_in, const int* in_sizes, int n_in,
                              void* d_out, int out_size, void* d_ws, size_t ws_size,
                              hipStream_t stream) {
  const float* x = (const float*)d_in[0];  // (4, 4096, 1024) f32
  const float* t = (const float*)d_in[1];  // (4096, 1024) f32
  float* y = (float*)d_out;                // (4, 4096, 1024) f32
  (void)in_sizes; (void)n_in; (void)out_size; (void)d_ws; (void)ws_size;

  dim3 grid(kD);     // one block per channel
  dim3 block(256);   // 8 wave32 waves
  toeplitz_causal_wmma<<<grid, block, 0, stream>>>(x, t, y);
}